// Router_15161234555446
// MI455X (gfx1250) — compile-verified
//
#include <hip/hip_runtime.h>

typedef __attribute__((ext_vector_type(2))) float v2f;
typedef __attribute__((ext_vector_type(4))) float v4f;
typedef __attribute__((ext_vector_type(8))) float v8f;

#define T_TOK 8192
#define DDIM  768
#define NEXP  8
#define CAP   2048

// ---------------------------------------------------------------------------
// Kernel A: router logits via V_WMMA_F32_16X16X4_F32 + softmax top-1.
// One wave (32 threads) per 16-token tile.
//   A (16x4 f32): lanes 0-15 -> M=0..15; VGPR0 = K { 0 | 2 }, VGPR1 = K { 1 | 3 }
//                 (lane half selects the +2 K offset) -> per-lane float2 load.
//   B (4x16 f32): lanes 0-15 hold N=0..15 of rows K { 0 | 2 } / { 1 | 3 };
//                 only N<8 is a real expert column. Lanes with N>=8 load a
//                 clamped-but-valid W address and select 0 via v_cndmask so
//                 no EXEC-masked branches appear in the inner loop.
//   D (16x16 f32, 8 VGPRs): VGPR j = row (j + 8*half), col = lane&15.
// D is staged through LDS so lanes 0-15 each own one token's 8 logits.
// ---------------------------------------------------------------------------
__global__ __launch_bounds__(32)
void router_logits_kernel(const float* __restrict__ x,
                          const float* __restrict__ W,
                          float* __restrict__ gate_out,
                          int*   __restrict__ idx_out) {
    __shared__ float lds[16][16];

    const int lane = threadIdx.x;
    const int m    = lane & 15;        // A row / B col
    const int half = lane >> 4;        // K sub-offset selector
    const int t0   = blockIdx.x * 16;
    const int ncol = m & 7;            // clamped (always-valid) W column
    const bool valid = (m < NEXP);     // hoisted predicate -> v_cndmask in loop

    const float* xrow = x + (size_t)(t0 + m) * DDIM;

    v8f c = {0.f, 0.f, 0.f, 0.f, 0.f, 0.f, 0.f, 0.f};

    #pragma unroll 4
    for (int k0 = 0; k0 < DDIM; k0 += 4) {
        const int kk = k0 + 2 * half;           // this lane's K pair
        v2f a = *(const v2f*)(xrow + kk);       // x[t0+m][kk], x[t0+m][kk+1]
        // Unconditional loads from valid addresses; zero-select afterwards.
        const float bx = W[(size_t)kk * NEXP + ncol];
        const float by = W[(size_t)(kk + 1) * NEXP + ncol];
        v2f b;
        b.x = valid ? bx : 0.0f;
        b.y = valid ? by : 0.0f;
        // 8 args: (neg_a, A, neg_b, B, c_mod, C, reuse_a, reuse_b)
        c = __builtin_amdgcn_wmma_f32_16x16x4_f32(false, a, false, b,
                                                  (short)0, c, false, false);
    }

    #pragma unroll
    for (int j = 0; j < 8; ++j)
        lds[j + 8 * half][m] = c[j];
    __syncthreads();

    if (lane < 16) {
        const int t = t0 + lane;
        float l[8];
        #pragma unroll
        for (int e = 0; e < 8; ++e) l[e] = lds[lane][e];

        float mx = l[0]; int ai = 0;
        #pragma unroll
        for (int e = 1; e < 8; ++e)
            if (l[e] > mx) { mx = l[e]; ai = e; }   // strict > keeps first (jax tie-break)

        float s = 0.0f;
        #pragma unroll
        for (int e = 0; e < 8; ++e) s += __expf(l[e] - mx);

        gate_out[t] = 1.0f / s;   // softmax prob of the argmax
        idx_out[t]  = ai;
    }
}

// ---------------------------------------------------------------------------
// Kernel B: sequential per-expert position (cumsum) + capacity mask.
// Single wave; each iteration assigns positions to 32 tokens using
// ballot/popcount prefix counting. Positions are 1-indexed (matches the
// reference: position = cumsum including self); kept iff pos < CAP.
// ---------------------------------------------------------------------------
__global__ __launch_bounds__(32)
void scan_kernel(const int* __restrict__ idx,
                 const float* __restrict__ gate,
                 int*   __restrict__ pos_out,
                 float* __restrict__ gf_out) {
    const int lane = threadIdx.x;
    const unsigned lower = (1u << lane) - 1u;

    int cnt[8];
    #pragma unroll
    for (int e = 0; e < 8; ++e) cnt[e] = 0;

    for (int base = 0; base < T_TOK; base += 32) {
        const int   t = base + lane;
        const int   e = idx[t];
        const float g = gate[t];
        int pos = 0;
        #pragma unroll
        for (int ex = 0; ex < 8; ++ex) {
            const unsigned msk = __builtin_amdgcn_ballot_w32(e == ex);
            if (e == ex)
                pos = cnt[ex] + __builtin_popcount(msk & lower) + 1;
            cnt[ex] += __builtin_popcount(msk);   // uniform across lanes
        }
        pos_out[t] = pos;
        gf_out[t]  = (pos < CAP) ? g : 0.0f;      // capacity-dropped -> gate 0
    }
}

// ---------------------------------------------------------------------------
// Kernel C: materialize dispatch (T,E,C) then combined (T,E,C), concatenated.
// One block per (t,e) row of 2048 floats in BOTH tensors; token scalars are
// loaded once per block. 1 GiB of output >> 192 MB L2 -> non-temporal stores.
// ---------------------------------------------------------------------------
__global__ __launch_bounds__(256)
void fill_kernel(const int* __restrict__ idx,
                 const int* __restrict__ pos,
                 const float* __restrict__ gf,
                 float* __restrict__ out) {
    const int bid = blockIdx.x;         // bid = t*8 + e
    const int t   = bid >> 3;
    const int e   = bid & 7;

    const int   s_idx = idx[t];
    const int   s_pos = pos[t];
    const float s_g   = gf[t];
    const bool  hot   = (e == s_idx) && (s_g != 0.0f);

    const size_t row = (size_t)bid * CAP;
    float* disp = out + row;
    float* comb = out + (size_t)T_TOK * NEXP * CAP + row;

    const int c0 = threadIdx.x * 8;

    v4f d0 = {0.f,0.f,0.f,0.f}, d1 = {0.f,0.f,0.f,0.f};
    v4f m0 = {0.f,0.f,0.f,0.f}, m1 = {0.f,0.f,0.f,0.f};

    if (hot && s_pos >= c0 && s_pos < c0 + 8) {
        const int r = s_pos - c0;
        if (r < 4) { d0[r]     = 1.0f; m0[r]     = s_g; }
        else       { d1[r - 4] = 1.0f; m1[r - 4] = s_g; }
    }

    __builtin_nontemporal_store(d0, (v4f*)(disp + c0));
    __builtin_nontemporal_store(d1, (v4f*)(disp + c0 + 4));
    __builtin_nontemporal_store(m0, (v4f*)(comb + c0));
    __builtin_nontemporal_store(m1, (v4f*)(comb + c0 + 4));
}

// ---------------------------------------------------------------------------
extern "C" void kernel_launch(void* const* d_in, const int* in_sizes, int n_in,
                              void* d_out, int out_size, void* d_ws, size_t ws_size,
                              hipStream_t stream) {
    const float* x = (const float*)d_in[0];   // (8192, 768) f32
    const float* W = (const float*)d_in[1];   // (768, 8)   f32
    float* out = (float*)d_out;               // dispatch ++ combined, f32

    char* ws = (char*)d_ws;                   // 128 KB of scratch used
    int*   ws_idx  = (int*)(ws + 0);          // 8192 int
    float* ws_gate = (float*)(ws + 32768);    // 8192 f32 (raw top-1 prob)
    int*   ws_pos  = (int*)(ws + 65536);      // 8192 int (1-indexed position)
    float* ws_gf   = (float*)(ws + 98304);    // 8192 f32 (capacity-masked gate)

    router_logits_kernel<<<T_TOK / 16, 32, 0, stream>>>(x, W, ws_gate, ws_idx);
    scan_kernel<<<1, 32, 0, stream>>>(ws_idx, ws_gate, ws_pos, ws_gf);
    fill_kernel<<<T_TOK * NEXP, 256, 0, stream>>>(ws_idx, ws_pos, ws_gf, out);
}